// MPSTPCP_51814485459242
// MI455X (gfx1250) — compile-verified
//
#include <hip/hip_runtime.h>

typedef float v2f __attribute__((ext_vector_type(2)));
typedef float v8f __attribute__((ext_vector_type(8)));

#define NQ     64
#define LAYERS 63
#define BATCH  65536

// SWAPX16: bitmask-mode ds_swizzle, and=0x1F, or=0, xor=0x10 -> 0x401F
__device__ __forceinline__ float swap16(float x) {
    return __int_as_float(__builtin_amdgcn_ds_swizzle(__float_as_int(x), 0x401F));
}

// Fast v_rcp_f32 (1 ulp) -- fine for density-matrix normalization, keeps the
// serial per-layer dependency chain short (TRANS op co-executes with WMMA).
__device__ __forceinline__ float fast_rcp(float x) {
    return __builtin_amdgcn_rcpf(x);
}

// Build per-layer 16x16 channel matrices:
// M[l][4a+d][4b+c] = sum_k K[l,k,a,b] * K[l,k,d,c]
__global__ void MPSTPCP_prep_kernel(const float* __restrict__ kraus,
                                    float* __restrict__ M) {
    const int l   = blockIdx.x;       // 0..62
    const int t   = threadIdx.x;      // 0..255
    const int row = t >> 4;           // 4a+d
    const int col = t & 15;           // 4b+c
    const int a = row >> 2, d = row & 3;
    const int b = col >> 2, c = col & 3;
    const float* kl = kraus + l * 64; // (K=4, 4, 4) block
    float s = 0.f;
#pragma unroll
    for (int k = 0; k < 4; ++k)
        s += kl[k * 16 + a * 4 + b] * kl[k * 16 + d * 4 + c];
    M[l * 256 + t] = s;
}

// Main kernel: one wave32 handles 16 batch elements (one 16x16 WMMA tile).
// State vec(rho) (16 comps) = M dim, batch = N dim.
// C/D layout: VGPR r, lanes 0-15 -> comp r, lanes 16-31 -> comp r+8.
__global__ __launch_bounds__(256, 4)
void MPSTPCP_main_kernel(const float* __restrict__ X,
                         const float* __restrict__ M,
                         float* __restrict__ out) {
    const int tid  = blockIdx.x * blockDim.x + threadIdx.x;
    const int wave = tid >> 5;
    const int lane = threadIdx.x & 31;
    const int n    = lane & 15;
    const int hi   = lane >> 4;
    const bool low = (hi == 0);
    const int batch = wave * 16 + n;
    const float* xb = X + (size_t)batch * (NQ * 2);

    float c0, c1, c2, c3, c4, c5, c6, c7;

    // --- initial state: rho = tensor(rho_q0, rho_q1) ---
    {
        float x0 = xb[0], x1 = xb[1];
        float inv = fast_rcp(x0 * x0 + x1 * x1);
        float w  = (hi ? x1 : x0) * inv;  // q0[i=hi, j] scale
        float pa = w * x0;                // q0[hi,0]
        float pb = w * x1;                // q0[hi,1]
        float y0 = xb[2], y1 = xb[3];
        float iv = fast_rcp(y0 * y0 + y1 * y1);
        pa *= iv; pb *= iv;
        float n00 = y0 * y0, n01 = y0 * y1, n11 = y1 * y1;
        c0 = pa * n00; c1 = pa * n01; c2 = pb * n00; c3 = pb * n01;
        c4 = pa * n01; c5 = pa * n11; c6 = pb * n01; c7 = pb * n11;
    }

    for (int l = 0; l < LAYERS; ++l) {
        // A fragments: M_l chunk j is 16x4, lane m holds (K=4j+2*hi, 4j+2*hi+1)
        const float* Ml = M + l * 256 + n * 16 + 2 * hi;
        v2f a0 = *(const v2f*)(Ml + 0);
        v2f a1 = *(const v2f*)(Ml + 4);
        v2f a2 = *(const v2f*)(Ml + 8);
        v2f a3 = *(const v2f*)(Ml + 12);

        // B fragments: 4x16 row chunks of state. VGPR0 = rows {4j, 4j+2},
        // VGPR1 = rows {4j+1, 4j+3} (low/high lane halves).
        float s0 = swap16(c0), s1 = swap16(c1), s2 = swap16(c2), s3 = swap16(c3);
        float s4 = swap16(c4), s5 = swap16(c5), s6 = swap16(c6), s7 = swap16(c7);
        v2f B0 = { low ? c0 : s2, low ? c1 : s3 };  // rows 0,2 / 1,3
        v2f B1 = { low ? c4 : s6, low ? c5 : s7 };  // rows 4,6 / 5,7
        v2f B2 = { low ? s0 : c2, low ? s1 : c3 };  // rows 8,10 / 9,11
        v2f B3 = { low ? s4 : c6, low ? s5 : c7 };  // rows 12,14 / 13,15

        v8f acc = {0.f, 0.f, 0.f, 0.f, 0.f, 0.f, 0.f, 0.f};
        acc = __builtin_amdgcn_wmma_f32_16x16x4_f32(false, a0, false, B0,
                                                    (short)0, acc, false, false);
        acc = __builtin_amdgcn_wmma_f32_16x16x4_f32(false, a1, false, B1,
                                                    (short)0, acc, false, false);
        acc = __builtin_amdgcn_wmma_f32_16x16x4_f32(false, a2, false, B2,
                                                    (short)0, acc, false, false);
        acc = __builtin_amdgcn_wmma_f32_16x16x4_f32(false, a3, false, B3,
                                                    (short)0, acc, false, false);

        if (l < LAYERS - 1) {
            // ptrace0: p[c,d] = v[4c+d] + v[4c+d+10]
            // low lanes need p0j, high lanes need p1j
            float t1 = swap16(low ? acc[4] : acc[2]);   // send v4 up / v10 down
            float pa = (low ? acc[0] : acc[6]) + t1;     // p00 | p10
            float t2 = swap16(low ? acc[5] : acc[3]);   // send v5 up / v11 down
            float pb = (low ? acc[1] : acc[7]) + t2;     // p01 | p11

            // attach next qubit: rho_new[2i+k,2j+l'] = p[i,j]*nxt[k,l']
            float q0 = xb[(l + 2) * 2], q1 = xb[(l + 2) * 2 + 1];
            float iv = fast_rcp(q0 * q0 + q1 * q1);
            pa *= iv; pb *= iv;
            float n00 = q0 * q0, n01 = q0 * q1, n11 = q1 * q1;
            c0 = pa * n00; c1 = pa * n01; c2 = pb * n00; c3 = pb * n01;
            c4 = pa * n01; c5 = pa * n11; c6 = pb * n01; c7 = pb * n11;
        } else {
            // result = ptrace0(rho)[0,0] = v[0] + v[10]
            float t = swap16(acc[2]);      // low lanes receive partner v10
            if (low) out[batch] = acc[0] + t;
        }
    }
}

extern "C" void kernel_launch(void* const* d_in, const int* in_sizes, int n_in,
                              void* d_out, int out_size, void* d_ws, size_t ws_size,
                              hipStream_t stream) {
    const float* X     = (const float*)d_in[0];   // (B, 64, 2) f32
    const float* kraus = (const float*)d_in[1];   // (63, 16, 4) f32
    float* out = (float*)d_out;                   // (B,) f32
    float* M   = (float*)d_ws;                    // 63*256 f32 = 63 KB

    MPSTPCP_prep_kernel<<<LAYERS, 256, 0, stream>>>(kraus, M);

    // 4096 tiles of 16 batch elems; 8 waves/block -> 512 blocks
    MPSTPCP_main_kernel<<<BATCH / 16 / 8, 256, 0, stream>>>(X, M, out);
}